// Fs_56616258895922
// MI455X (gfx1250) — compile-verified
//
#include <hip/hip_runtime.h>

typedef __attribute__((ext_vector_type(2))) float v2f;
typedef __attribute__((ext_vector_type(8))) float v8f;

#define WAVES_PER_BLOCK 4   // waves per workgroup
#define RT 2                // row tiles (of 16 rows) per wave

// Raw CDNA5 min ops: avoid LLVM's sNaN-canonicalization (v_max v,v,v) around
// fminf. v_min_num/v_min3_num are IEEE minimumNumber in hardware.
__device__ __forceinline__ float min2f(float a, float b) {
    float r;
    asm("v_min_num_f32 %0, %1, %2" : "=v"(r) : "v"(a), "v"(b));
    return r;
}
__device__ __forceinline__ float min3f(float a, float b, float c) {
    float r;
    asm("v_min3_num_f32 %0, %1, %2, %3" : "=v"(r) : "v"(a), "v"(b), "v"(c));
    return r;
}

// ---------------------------------------------------------------------------
// Pack (p0,p1,p2) -> (p0, p1, p2, ||p||^2) as float4 (one b128 store/point).
// Gives branch-free WMMA operand loads in the hot kernel.
// ---------------------------------------------------------------------------
__global__ __launch_bounds__(256)
void chamfer_pack_kernel(const float* __restrict__ in,   // (total, 3)
                         float4* __restrict__ out,       // (total,)
                         int total)
{
    int i = blockIdx.x * blockDim.x + threadIdx.x;
    if (i >= total) return;
    const float p0 = in[(size_t)i * 3 + 0];
    const float p1 = in[(size_t)i * 3 + 1];
    const float p2 = in[(size_t)i * 3 + 2];
    float4 q;
    q.x = p0; q.y = p1; q.z = p2;
    q.w = p0 * p0 + p1 * p1 + p2 * p2;
    out[i] = q;
}

// ---------------------------------------------------------------------------
// One wave owns RT*16 "row" points (X) and sweeps ALL "col" points (Y),
// evaluating 16x16 squared-distance tiles with V_WMMA_F32_16X16X4_F32:
//   A(16x4)  row m : (-2x0, -2x1, -2x2, 1)
//   B(4x16)  col n : ( y0,   y1,   y2,  ||y||^2)
//   C[m][n]        : ||x_m||^2
//   D = A*B + C    : squared distance tile
// Two column tiles per iteration -> rmin = v_min3(rmin, d0, d1).
// ---------------------------------------------------------------------------
__global__ __launch_bounds__(32 * WAVES_PER_BLOCK)
void chamfer_rowmin_kernel(const float4* __restrict__ Xq,  // (B, N) packed
                           const float4* __restrict__ Yq,  // (B, M) packed
                           float* __restrict__ rowmin,     // (B, N)
                           int N, int M)
{
    const int lane = threadIdx.x & 31;
    const int wave = threadIdx.x >> 5;
    const int b    = blockIdx.y;
    const int rowTiles = (N + 15) >> 4;
    const int group = blockIdx.x * WAVES_PER_BLOCK + wave;  // group of RT tiles
    const int tile0 = group * RT;
    if (tile0 >= rowTiles) return;                          // whole-wave exit

    const int  l15 = lane & 15;
    const bool hi  = lane >= 16;
    const int  cb  = hi ? 8 : 0;     // C/D: lanes 16-31 hold rows 8..15

    const float4* Xb = Xq + (size_t)b * N;

    v2f a[RT];
    v8f c[RT];
    v8f rmin[RT];

    #pragma unroll
    for (int r = 0; r < RT; ++r) {
        int tr  = tile0 + r; if (tr >= rowTiles) tr = rowTiles - 1;
        int row = (tr << 4) + l15; if (row >= N) row = N - 1;
        const float4 q = Xb[row];
        if (!hi) { a[r].x = -2.0f * q.x; a[r].y = -2.0f * q.y; }  // K=0,1
        else     { a[r].x = -2.0f * q.z; a[r].y = 1.0f;        }  // K=2,3
        const float xn = q.w;   // lane l holds ||x||^2 of row tr*16+l
        #pragma unroll
        for (int v = 0; v < 8; ++v)
            c[r][v] = __shfl(xn, cb + v, 32);
        #pragma unroll
        for (int v = 0; v < 8; ++v) rmin[r][v] = 3.0e38f;
    }

    // B operand stream: packed points viewed as v2f pairs.
    // lo lanes read entry 2*col (y0,y1); hi lanes entry 2*col+1 (y2,||y||^2).
    const v2f* Yb2 = (const v2f*)(Yq + (size_t)b * M);
    int idx = (l15 << 1) | (hi ? 1 : 0);

    const int fullTiles = M >> 4;
    const int pairTiles = fullTiles >> 1;

    #pragma unroll 2
    for (int t = 0; t < pairTiles; ++t) {
        const v2f bb0 = Yb2[idx];
        const v2f bb1 = Yb2[idx + 32];
        idx += 64;                                   // 2 tiles * 16 cols * 2
        #pragma unroll
        for (int r = 0; r < RT; ++r) {
            v8f d0 = __builtin_amdgcn_wmma_f32_16x16x4_f32(
                false, a[r], false, bb0, (short)0, c[r], false, false);
            v8f d1 = __builtin_amdgcn_wmma_f32_16x16x4_f32(
                false, a[r], false, bb1, (short)0, c[r], false, false);
            #pragma unroll
            for (int v = 0; v < 8; ++v)
                rmin[r][v] = min3f(rmin[r][v], d0[v], d1[v]);
        }
    }
    if (fullTiles & 1) {                             // odd full tile
        const v2f bb = Yb2[idx];
        idx += 32;
        #pragma unroll
        for (int r = 0; r < RT; ++r) {
            v8f d = __builtin_amdgcn_wmma_f32_16x16x4_f32(
                false, a[r], false, bb, (short)0, c[r], false, false);
            #pragma unroll
            for (int v = 0; v < 8; ++v) rmin[r][v] = min2f(rmin[r][v], d[v]);
        }
    }
    if (M & 15) {                                    // clamped tail tile
        int col = (fullTiles << 4) + l15; if (col >= M) col = M - 1;
        const v2f bb = Yb2[(col << 1) | (hi ? 1 : 0)];
        #pragma unroll
        for (int r = 0; r < RT; ++r) {
            v8f d = __builtin_amdgcn_wmma_f32_16x16x4_f32(
                false, a[r], false, bb, (short)0, c[r], false, false);
            #pragma unroll
            for (int v = 0; v < 8; ++v) rmin[r][v] = min2f(rmin[r][v], d[v]);
        }
    }

    // Min over the 16 columns: butterfly stays inside each half-wave
    // (each half holds all 16 columns of its 8 rows).
    #pragma unroll
    for (int off = 1; off < 16; off <<= 1) {
        #pragma unroll
        for (int r = 0; r < RT; ++r)
            #pragma unroll
            for (int v = 0; v < 8; ++v)
                rmin[r][v] = min2f(rmin[r][v], __shfl_xor(rmin[r][v], off, 32));
    }

    if (l15 == 0) {
        #pragma unroll
        for (int r = 0; r < RT; ++r) {
            int tr = tile0 + r;
            if (tr >= rowTiles) break;
            const int mbase = (tr << 4) + (hi ? 8 : 0);
            #pragma unroll
            for (int v = 0; v < 8; ++v) {
                const int rr = mbase + v;
                if (rr < N) rowmin[(size_t)b * N + rr] = rmin[r][v];
            }
        }
    }
}

// ---------------------------------------------------------------------------
// One block per batch: threshold, mean over N and M, sum of the two means.
// ---------------------------------------------------------------------------
__global__ __launch_bounds__(256)
void chamfer_finalize_kernel(const float* __restrict__ d1,  // (B, N)
                             const float* __restrict__ d2,  // (B, M)
                             const float* __restrict__ thp,
                             float* __restrict__ out,       // (B,)
                             int N, int M)
{
    const int b   = blockIdx.x;
    const int tid = threadIdx.x;
    const float th = *thp;

    float s1 = 0.0f, s2 = 0.0f;
    for (int i = tid; i < N; i += blockDim.x) {
        float v = d1[(size_t)b * N + i];
        s1 += (v <= th) ? 0.0f : v;
    }
    for (int i = tid; i < M; i += blockDim.x) {
        float v = d2[(size_t)b * M + i];
        s2 += (v <= th) ? 0.0f : v;
    }
    float s = s1 / (float)N + s2 / (float)M;

    __shared__ float sm[256];
    sm[tid] = s;
    __syncthreads();
    for (int off = 128; off > 0; off >>= 1) {
        if (tid < off) sm[tid] += sm[tid + off];
        __syncthreads();
    }
    if (tid == 0) out[b] = sm[0];
}

extern "C" void kernel_launch(void* const* d_in, const int* in_sizes, int n_in,
                              void* d_out, int out_size, void* d_ws, size_t ws_size,
                              hipStream_t stream)
{
    const float* xyz1 = (const float*)d_in[0];   // (B, N, 3)
    const float* xyz2 = (const float*)d_in[1];   // (B, M, 3)
    const float* th   = (const float*)d_in[2];   // scalar
    float* out = (float*)d_out;

    const int B = out_size;                      // reference returns (B,)
    const int N = in_sizes[0] / (3 * B);
    const int M = in_sizes[1] / (3 * B);

    // Workspace layout
    float*  d1min = (float*)d_ws;                          // B*N floats
    float*  d2min = d1min + (size_t)B * N;                 // B*M floats
    float4* X4    = (float4*)(d2min + (size_t)B * M);      // B*N float4
    float4* Y4    = X4 + (size_t)B * N;                    // B*M float4

    // Pack both clouds into (p, ||p||^2) float4 form.
    {
        const int tX = B * N, tY = B * M;
        chamfer_pack_kernel<<<(tX + 255) / 256, 256, 0, stream>>>(xyz1, X4, tX);
        chamfer_pack_kernel<<<(tY + 255) / 256, 256, 0, stream>>>(xyz2, Y4, tY);
    }

    // d1: for each x, min over all y
    {
        const int rowTiles = (N + 15) >> 4;
        const int groups   = (rowTiles + RT - 1) / RT;
        dim3 grid((groups + WAVES_PER_BLOCK - 1) / WAVES_PER_BLOCK, B);
        chamfer_rowmin_kernel<<<grid, 32 * WAVES_PER_BLOCK, 0, stream>>>(
            X4, Y4, d1min, N, M);
    }
    // d2: for each y, min over all x (swap roles instead of atomics)
    {
        const int rowTiles = (M + 15) >> 4;
        const int groups   = (rowTiles + RT - 1) / RT;
        dim3 grid((groups + WAVES_PER_BLOCK - 1) / WAVES_PER_BLOCK, B);
        chamfer_rowmin_kernel<<<grid, 32 * WAVES_PER_BLOCK, 0, stream>>>(
            Y4, X4, d2min, M, N);
    }

    chamfer_finalize_kernel<<<B, 256, 0, stream>>>(d1min, d2min, th, out, N, M);
}